// CausalSegmentationSetHash_5119601017066
// MI455X (gfx1250) — compile-verified
//
#include <hip/hip_runtime.h>
#include <hip/hip_bf16.h>

// ---------------------------------------------------------------------------
// CausalSegmentation forward on MI455X (gfx1250, wave32).
//
// Algebraic restructuring: seg_mean @ phi_w = ((pref[t]-pref[t-L])/(L+8)) @ phi_w
//   = (P[t]-P[t-L])/(L+8) with P = cumsum(u @ phi_w)  -> removes the per-step
// (B*lmax)x256x128 GEMM (~69 GFLOP); remaining GEMMs (~4 GFLOP total) use fp32
// WMMA with 16x128 register blocking (A fragment reused across 8 WMMAs).
// ---------------------------------------------------------------------------

#define BB 8
#define TT 2048
#define CC 128
#define DR 256
#define LMAX 64
#define SS 4

typedef __attribute__((ext_vector_type(2))) float v2f;
typedef __attribute__((ext_vector_type(8))) float v8f;

// ---------------------------------------------------------------------------
// fp32 WMMA GEMM: out(M x 128) = A(M x K) @ Bm(K x 128) + epilogue.
// One wave computes a full 16(M) x 128(N) stripe: 8 accumulators, A fragment
// loaded once per k-step (1 b64) and reused by 8 back-to-back WMMAs; B rows
// are tiny and block-shared -> L0/L2 hits.
// MODE 0: z -> RFF u (adds bias, writes [cos,sin]/sqrt(128) into width-256 out)
// MODE 1: plain store
// MODE 2: store + bias
// ---------------------------------------------------------------------------
template <int MODE>
__global__ __launch_bounds__(256) void wmma_gemm_f32_kernel(
    const float* __restrict__ A, const float* __restrict__ Bm,
    const float* __restrict__ bias, float* __restrict__ out, int M, int K) {
  const int N = 128;
  const int wave = blockIdx.x * (blockDim.x >> 5) + (threadIdx.x >> 5);
  const int lane = threadIdx.x & 31;
  const int m0 = wave * 16;
  if (m0 >= M) return;

  // A 16x4 fp32 layout: lanes 0-15 row M=lane hold K={k,k+1}; lanes 16-31 K={k+2,k+3}
  const int r15 = lane & 15;
  const int kg  = (lane >> 4) * 2;
  const float* Arow = A + (size_t)(m0 + r15) * (size_t)K;

  v8f acc[8];
#pragma unroll
  for (int nt = 0; nt < 8; ++nt) acc[nt] = (v8f){};

  for (int k = 0; k < K; k += 4) {
    v2f a;
    a.x = Arow[k + kg];
    a.y = Arow[k + kg + 1];
    // B 4x16 fp32 layout mirrors A: lanes 0-15 hold K={k,k+1} at N=col, lanes 16-31 K={k+2,k+3}
    const float* B0 = Bm + (size_t)(k + kg) * N + r15;
    const float* B1 = B0 + N;
#pragma unroll
    for (int nt = 0; nt < 8; ++nt) {
      v2f b;
      b.x = B0[nt * 16];
      b.y = B1[nt * 16];
      acc[nt] = __builtin_amdgcn_wmma_f32_16x16x4_f32(false, a, false, b,
                                                      (short)0, acc[nt], false, false);
    }
  }

  // C/D layout: VGPR j -> row m0 + j + (lane>=16 ? 8 : 0), col n0 + (lane&15)
  const int rbase = m0 + (lane >> 4) * 8;
#pragma unroll
  for (int nt = 0; nt < 8; ++nt) {
    const int colD = nt * 16 + r15;
#pragma unroll
    for (int j = 0; j < 8; ++j) {
      const int row = rbase + j;
      float v = acc[nt][j];
      if (MODE == 0) {
        const float z = v + bias[colD];
        float sn, cs;
        __sincosf(z, &sn, &cs);
        const float inv = 0.08838834764831845f;  // 1/sqrt(128)
        out[(size_t)row * 256 + colD]       = cs * inv;
        out[(size_t)row * 256 + 128 + colD] = sn * inv;
      } else if (MODE == 1) {
        out[(size_t)row * 128 + colD] = v;
      } else {
        out[(size_t)row * 128 + colD] = v + bias[colD];
      }
    }
  }
}

// ---------------------------------------------------------------------------
// P[b,0,:]=0 ; P[b,t,:] = P[b,t-1,:] + Pu[b,t-1,:]   (per-channel causal cumsum)
// 1024 independent series (b,c); threads consecutive in c -> coalesced.
// global_prefetch_b8 ahead of the serial stream to hide latency.
// ---------------------------------------------------------------------------
__global__ void cumsum_kernel(const float* __restrict__ Pu,  // (B,T,C)
                              float* __restrict__ P) {       // (B,T+1,C)
  const int idx = blockIdx.x * blockDim.x + threadIdx.x;     // 0..B*C-1
  const int b = idx >> 7;
  const int c = idx & (CC - 1);
  if (b >= BB) return;
  const float* src = Pu + (size_t)b * TT * CC + c;
  float* dst = P + (size_t)b * (TT + 1) * CC + c;
  float run = 0.0f;
  dst[0] = 0.0f;
  for (int t = 1; t <= TT; ++t) {
    if (t + 16 <= TT) __builtin_prefetch(src + (size_t)(t + 15) * CC, 0, 0);
    run += src[(size_t)(t - 1) * CC];
    dst[(size_t)t * CC] = run;
  }
}

// ---------------------------------------------------------------------------
// Sequential scan. One block per (s,b); LDS ring buffers for logZ/q/k over the
// last 64 steps. Wave 0 does the length-softmax with wave32 shuffles; all 8
// waves build the 64x128 tanh segment tile; both 128-thread halves split the
// q mix (serial chain halved to 32 iters) with a partial-sum combine.
// Writes h[b, t-1, s*128 + c] = rep.
// ---------------------------------------------------------------------------
__global__ __launch_bounds__(256) void recurrence_kernel(
    const float* __restrict__ P,       // (B,T+1,C)
    const float* __restrict__ phi_b,   // (C)
    const float* __restrict__ log_w,   // (S,LMAX)
    const float* __restrict__ anchor,  // (S,C)
    float* __restrict__ h) {           // (B,T,S*C)
  extern __shared__ float smem[];
  float* sh_seg  = smem;                    // LMAX*CC
  float* sh_q    = sh_seg + LMAX * CC;      // LMAX*CC
  float* sh_part = sh_q + LMAX * CC;        // 256 partial sums
  float* sh_logZ = sh_part + 256;           // LMAX
  float* sh_k    = sh_logZ + LMAX;          // LMAX
  float* sh_pi   = sh_k + LMAX;             // LMAX
  float* sh_s    = sh_pi + LMAX;            // [0]=logZ_t [1]=k_t

  const int tid  = threadIdx.x;
  const int sidx = blockIdx.x & 3;
  const int b    = blockIdx.x >> 2;

  for (int i = tid; i < LMAX * CC; i += blockDim.x) sh_q[i] = 0.0f;
  if (tid < LMAX) {
    sh_logZ[tid] = (tid == 0) ? 0.0f : -1e30f;  // logZ[0]=0, j<0 -> masked
    sh_k[tid] = 0.0f;
  }
  __syncthreads();

  const int cc = tid & (CC - 1);
  const int lh = tid >> 7;  // 0 or 1: thread-half
  const float phib_c = phi_b[cc];
  const float anc    = anchor[sidx * CC + cc];
  const float* Pb    = P + (size_t)b * (TT + 1) * CC;
  const float* lw    = log_w + sidx * LMAX;

  for (int t = 1; t <= TT; ++t) {
    // ---- Phase 1 (wave 0): length softmax pi, k_t, logZ_t ----
    if (tid < 32) {
      const int l0 = tid, l1 = tid + 32;
      const int j0 = t - (l0 + 1), j1 = t - (l1 + 1);
      const float la0 = lw[l0] + sh_logZ[j0 & 63];
      const float la1 = lw[l1] + sh_logZ[j1 & 63];
      float mx = fmaxf(la0, la1);
#pragma unroll
      for (int off = 16; off > 0; off >>= 1) mx = fmaxf(mx, __shfl_xor(mx, off, 32));
      const float e0 = __expf(la0 - mx);
      const float e1 = __expf(la1 - mx);
      float sum = e0 + e1;
#pragma unroll
      for (int off = 16; off > 0; off >>= 1) sum += __shfl_xor(sum, off, 32);
      const float inv = 1.0f / sum;
      const float p0 = e0 * inv, p1 = e1 * inv;
      sh_pi[l0] = p0;
      sh_pi[l1] = p1;
      float kt = p0 * (sh_k[j0 & 63] + 1.0f) + p1 * (sh_k[j1 & 63] + 1.0f);
#pragma unroll
      for (int off = 16; off > 0; off >>= 1) kt += __shfl_xor(kt, off, 32);
      if (tid == 0) {
        sh_s[0] = mx + __logf(sum);  // logsumexp
        sh_s[1] = kt;
      }
    }
    // ---- Phase 2 (all 256 threads): seg tile tanh((P[t]-P[t-L])/(L+8)+phi_b) ----
    const float Ptc = Pb[(size_t)t * CC + cc];
    for (int li = lh; li < LMAX; li += 2) {
      const int L = li + 1;
      int j = t - L;
      if (j < 0) j = 0;  // masked by pi==0; clamp keeps values finite
      const float v = (Ptc - Pb[(size_t)j * CC + cc]) / (float)(L + 8) + phib_c;
      sh_seg[li * CC + cc] = tanhf(v);
    }
    __syncthreads();

    // ---- Phase 3: q mix, split l-range across thread halves ----
    {
      const int lbeg = lh * 32, lend = lbeg + 32;
      float part = 0.0f;
#pragma unroll 4
      for (int li = lbeg; li < lend; ++li) {
        const int j = t - (li + 1);
        part += sh_pi[li] * (sh_q[(j & 63) * CC + cc] + sh_seg[li * CC + cc]);
      }
      sh_part[tid] = part;
    }
    __syncthreads();

    // ---- Phase 4: combine, ring update + rep output ----
    const int slot = t & 63;
    const float kt = sh_s[1];
    if (tid < CC) {
      const float qt = sh_part[tid] + sh_part[tid + 128];
      sh_q[slot * CC + cc] = qt;
      const float rep = (qt + 16.0f * anc) / (kt + 16.0f) * (kt / (kt + 16.0f));
      h[(size_t)(b * TT + (t - 1)) * (SS * CC) + sidx * CC + cc] = rep;
    }
    if (tid == CC) {
      sh_logZ[slot] = sh_s[0];
      sh_k[slot] = kt;
    }
    __syncthreads();
  }
}

// ---------------------------------------------------------------------------
extern "C" void kernel_launch(void* const* d_in, const int* in_sizes, int n_in,
                              void* d_out, int out_size, void* d_ws, size_t ws_size,
                              hipStream_t stream) {
  const float* x      = (const float*)d_in[0];  // (B,T,C)
  const float* W      = (const float*)d_in[1];  // (C,128)
  const float* rffb   = (const float*)d_in[2];  // (128)
  const float* phi_w  = (const float*)d_in[3];  // (256,C)
  const float* phi_b  = (const float*)d_in[4];  // (C)
  const float* anchor = (const float*)d_in[5];  // (S,C)
  const float* log_w  = (const float*)d_in[6];  // (S,LMAX)
  const float* mix_w  = (const float*)d_in[7];  // (S*C,C)
  const float* mix_b  = (const float*)d_in[8];  // (C)
  float* out = (float*)d_out;                   // (B,T,C)

  const size_t M = (size_t)BB * TT;  // 16384 rows
  float* ws = (float*)d_ws;
  float* u  = ws;                                   // M*256
  float* Pu = u + M * 256;                          // M*128
  float* P  = Pu + M * 128;                         // B*(T+1)*128
  float* h  = P + (size_t)BB * (TT + 1) * 128;      // M*512

  const dim3 blk(256);
  const int gemm_blocks = (int)(M / 16 / 8);  // 1024 waves, 8 waves per block

  // 1) z = x@W + b ; u = [cos z, sin z]/sqrt(128)
  wmma_gemm_f32_kernel<0><<<gemm_blocks, blk, 0, stream>>>(x, W, rffb, u, (int)M, CC);
  // 2) Pu = u @ phi_w
  wmma_gemm_f32_kernel<1><<<gemm_blocks, blk, 0, stream>>>(u, phi_w, nullptr, Pu, (int)M, DR);
  // 3) P = causal cumsum of Pu (zero-prepended)
  cumsum_kernel<<<(BB * CC + 255) / 256, 256, 0, stream>>>(Pu, P);
  // 4) sequential scan -> h (rep, laid out (B,T,S*C))
  const size_t shbytes = (size_t)(LMAX * CC * 2 + 256 + LMAX * 3 + 8) * sizeof(float);
  recurrence_kernel<<<SS * BB, blk, shbytes, stream>>>(P, phi_b, log_w, anchor, h);
  // 5) out = h @ mix_w + mix_b
  wmma_gemm_f32_kernel<2><<<gemm_blocks, blk, 0, stream>>>(h, mix_w, mix_b, out, (int)M, SS * CC);
}